// GroupStratifiedSurvivalLoss_88003879895270
// MI455X (gfx1250) — compile-verified
//
#include <hip/hip_runtime.h>
#include <hip/hip_bf16.h>

typedef __attribute__((ext_vector_type(2))) float v2f;
typedef __attribute__((ext_vector_type(8))) float v8f;

#define NGROUP 16
#define BLKTHREADS 256
#define BINSTRIDE 17   // pad 16 -> 17 words to spread LDS banks

// ---- per-sample Weibull NLL ----
// pow(x,b) = exp2(b * log2(x)) on the native v_exp_f32/v_log_f32 units.
__device__ __forceinline__ float weibull_nll(float b, float a, float y, float u) {
    float inv_a = __builtin_amdgcn_rcpf(a);          // v_rcp_f32
    float x0 = y + 1e-35f * inv_a;                   // == y in f32, kept for fidelity
    float x1 = y + inv_a;
    float h0 = __builtin_amdgcn_exp2f(b * __builtin_amdgcn_logf(x0)); // v_exp/v_log
    float h1 = __builtin_amdgcn_exp2f(b * __builtin_amdgcn_logf(x1));
    float t  = __expf(h1 - h0) - 1.0f;
    return h1 - u * __logf(t);                       // -(u*log(t) - h1)
}

// Kernel 1: streaming pass. Each thread owns 16 LDS sum slots + 16 LDS count
// slots (single writer per slot -> deterministic; same-wave DS ops are
// in-order). One ds_add_f32 per accumulation instead of a 48-op VALU chain.
// Block then tree-reduces the per-thread bins and emits per-block partials:
//   partial[blk*32 + g]      = group sums   (g in 0..15)
//   partial[blk*32 + 16 + g] = group counts (g in 0..15)
__global__ void __launch_bounds__(BLKTHREADS)
survloss_partial(const float* __restrict__ shp, const float* __restrict__ scl,
                 const float* __restrict__ tim, const float* __restrict__ evt,
                 const int* __restrict__ grp, float* __restrict__ partial,
                 int n, int nv) {
    __shared__ float lsum[BLKTHREADS * BINSTRIDE];
    __shared__ float lcnt[BLKTHREADS * BINSTRIDE];

    const int t     = threadIdx.x;
    const int tbase = t * BINSTRIDE;
#pragma unroll
    for (int g = 0; g < NGROUP; ++g) { lsum[tbase + g] = 0.0f; lcnt[tbase + g] = 0.0f; }
    // no barrier needed yet: each thread touches only its own slots

    const int tid    = blockIdx.x * blockDim.x + threadIdx.x;
    const int stride = gridDim.x * blockDim.x;

    const float4* b4 = (const float4*)shp;
    const float4* a4 = (const float4*)scl;
    const float4* y4 = (const float4*)tim;
    const float4* u4 = (const float4*)evt;
    const int4*   g4 = (const int4*)grp;

    for (int i = tid; i < nv; i += stride) {
        // forward prefetch of next strided chunk (global_prefetch_b8)
        __builtin_prefetch(&b4[i + stride], 0, 0);
        __builtin_prefetch(&a4[i + stride], 0, 0);
        __builtin_prefetch(&y4[i + stride], 0, 0);
        __builtin_prefetch(&u4[i + stride], 0, 0);
        __builtin_prefetch(&g4[i + stride], 0, 0);

        float4 bb = b4[i];
        float4 aa = a4[i];
        float4 yy = y4[i];
        float4 uu = u4[i];
        int4   gg = g4[i];

        float p0 = weibull_nll(bb.x, aa.x, yy.x, uu.x);
        float p1 = weibull_nll(bb.y, aa.y, yy.y, uu.y);
        float p2 = weibull_nll(bb.z, aa.z, yy.z, uu.z);
        float p3 = weibull_nll(bb.w, aa.w, yy.w, uu.w);

        atomicAdd(&lsum[tbase + gg.x], p0);  atomicAdd(&lcnt[tbase + gg.x], 1.0f);
        atomicAdd(&lsum[tbase + gg.y], p1);  atomicAdd(&lcnt[tbase + gg.y], 1.0f);
        atomicAdd(&lsum[tbase + gg.z], p2);  atomicAdd(&lcnt[tbase + gg.z], 1.0f);
        atomicAdd(&lsum[tbase + gg.w], p3);  atomicAdd(&lcnt[tbase + gg.w], 1.0f);
    }

    // scalar tail (n not multiple of 4): handled by lowest global threads
    int rem = n - (nv << 2);
    if (tid < rem) {
        int i = (nv << 2) + tid;
        float p = weibull_nll(shp[i], scl[i], tim[i], evt[i]);
        int g = grp[i];
        atomicAdd(&lsum[tbase + g], p);
        atomicAdd(&lcnt[tbase + g], 1.0f);
    }
    __syncthreads();

    // deterministic fixed-order tree reduction across threads
    for (int off = BLKTHREADS / 2; off >= 1; off >>= 1) {
        if (t < off) {
#pragma unroll
            for (int g = 0; g < NGROUP; ++g) {
                lsum[t * BINSTRIDE + g] += lsum[(t + off) * BINSTRIDE + g];
                lcnt[t * BINSTRIDE + g] += lcnt[(t + off) * BINSTRIDE + g];
            }
        }
        __syncthreads();
    }
    if (t < NGROUP) {
        partial[blockIdx.x * 32 + t]          = lsum[t];   // row 0 of lsum
        partial[blockIdx.x * 32 + NGROUP + t] = lcnt[t];   // row 0 of lcnt
    }
}

// Kernel 2: one wave32. Column-sum of [nb x 16] partials via
// V_WMMA_F32_16X16X4_F32 with an all-ones A matrix (ones[16x4] x P[4x16]).
// D row 0 (lanes 0..15, vgpr 0) = the 16 group totals. Deterministic, f32.
__global__ void __launch_bounds__(32)
survloss_finalize(const float* __restrict__ partial, float* __restrict__ out,
                  int nb, int n) {
    const int lane = threadIdx.x;        // 0..31, all active (EXEC all-1s)
    const int col  = lane & 15;          // N index
    const int row0 = (lane >> 4) * 2;    // K rows held by this lane group

    v2f ones; ones.x = 1.0f; ones.y = 1.0f;
    v8f accS = {};
    v8f accC = {};

    for (int k = 0; k < nb; k += 4) {
        v2f bS, bC;
        bS.x = partial[(k + row0 + 0) * 32 + col];
        bS.y = partial[(k + row0 + 1) * 32 + col];
        bC.x = partial[(k + row0 + 0) * 32 + NGROUP + col];
        bC.y = partial[(k + row0 + 1) * 32 + NGROUP + col];
        accS = __builtin_amdgcn_wmma_f32_16x16x4_f32(
            false, ones, false, bS, (short)0, accS, false, false);
        accC = __builtin_amdgcn_wmma_f32_16x16x4_f32(
            false, ones, false, bC, (short)0, accC, false, false);
    }

    __shared__ float gs[NGROUP], gc[NGROUP];
    if (lane < NGROUP) {
        gs[lane] = accS[0];   // D[0][lane] = column (group) sum
        gc[lane] = accC[0];
    }
    __syncthreads();

    if (lane == 0) {
        float tot = 0.0f;
        for (int g = 0; g < NGROUP; ++g) tot += gs[g];
        out[0] = tot / (float)n;                           // overall mean loss
    }
    if (lane < NGROUP) {
        out[1 + lane] = gs[lane] / fmaxf(gc[lane], 1.0f);  // per-group means
    }
}

extern "C" void kernel_launch(void* const* d_in, const int* in_sizes, int n_in,
                              void* d_out, int out_size, void* d_ws, size_t ws_size,
                              hipStream_t stream) {
    const float* shp = (const float*)d_in[0];  // shape
    const float* scl = (const float*)d_in[1];  // scale
    const float* tim = (const float*)d_in[2];  // time
    const float* evt = (const float*)d_in[3];  // event
    // d_in[4] = lengths (unused; only its length N matters)
    const int*   grp = (const int*)d_in[5];    // group ids in [0,16)

    float* out     = (float*)d_out;            // [0]=loss, [1..16]=group losses
    float* partial = (float*)d_ws;             // nb * 32 floats

    const int n  = in_sizes[0];
    const int nv = n >> 2;                     // float4 chunks

    int nb = 1024;                             // blocks; multiple of 4 for WMMA K
    while (nb > 4 && (size_t)nb * 32 * sizeof(float) > ws_size) nb >>= 1;

    survloss_partial<<<nb, BLKTHREADS, 0, stream>>>(shp, scl, tim, evt, grp,
                                                    partial, n, nv);
    survloss_finalize<<<1, 32, 0, stream>>>(partial, out, nb, n);
}